// GraphResNet_42872363549123
// MI455X (gfx1250) — compile-verified
//
#include <hip/hip_runtime.h>
#include <hip/hip_bf16.h>

#define GN 25000
#define GE 400000
#define GH 192

// Pair-swizzled LDS chunk: 8 k-groups x 2 lane-halves x (192 cols * 2 + 32 pad)
#define HSTRIDE 416                      // 384 data floats + 32 pad (bank shift)
#define SWSZ (16 * HSTRIDE)              // 6656 floats = 26624 B

typedef float v2f __attribute__((ext_vector_type(2)));
typedef float v8f __attribute__((ext_vector_type(8)));

// ---------------------------------------------------------------------------
// Dense GEMM: Y[N,192] = X[N,192] @ W[192,192] using V_WMMA_F32_16X16X4_F32.
// 8 waves/block, each wave computes one 16-row stripe across all 12 col tiles.
// W staged in pair-swizzled LDS chunks; all 12 B fragments of a k-step are
// loaded into distinct registers before the WMMA burst so DS waits batch.
// ---------------------------------------------------------------------------
__global__ __launch_bounds__(256) void gemm192_wmma(const float* __restrict__ X,
                                                    const float* __restrict__ W,
                                                    float* __restrict__ Y,
                                                    int nrows) {
  __shared__ float sW[SWSZ];
  const int wave = threadIdx.x >> 5;
  const int lane = threadIdx.x & 31;
  const int hi = lane >> 4;      // 0 for lanes 0-15, 1 for lanes 16-31
  const int lo = lane & 15;
  const int m0 = (blockIdx.x * 8 + wave) * 16;
  const int arow = m0 + lo;
  // Clamp: OOB A-rows only feed OOB D-rows, which are masked at the store.
  const float* xrow = X + (size_t)(arow < nrows ? arow : 0) * GH;

  v8f acc[12] = {};

  for (int kb = 0; kb < GH; kb += 32) {
    __syncthreads();
    // Stage W rows [kb, kb+32), pair-swizzled:
    //   sW[(kgrp*2 + half)*HSTRIDE + col*2 + pair] = W[kb + kgrp*4 + half*2 + pair][col]
    for (int i = threadIdx.x; i < 32 * GH; i += 256) {
      const int rr = i / GH;         // row within chunk (0..31)
      const int c = i - rr * GH;     // column (0..191)
      const int kgrp = rr >> 2;
      const int half = (rr >> 1) & 1;
      const int pair = rr & 1;
      sW[(kgrp * 2 + half) * HSTRIDE + c * 2 + pair] = W[kb * GH + i];
    }
    __syncthreads();

#pragma unroll
    for (int kk = 0; kk < 32; kk += 4) {
      // A fragment (16x4 f32): lanes 0-15 carry K=k,k+1; lanes 16-31 K=k+2,k+3
      const v2f a = *(const v2f*)(xrow + kb + kk + 2 * hi);
      const float* bbase = &sW[((kk >> 2) * 2 + hi) * HSTRIDE + lo * 2];
      // Batch-load all 12 B fragments into distinct registers first so the
      // scheduler can issue the DS loads together and burst the 12 WMMAs.
      v2f bf[12];
#pragma unroll
      for (int t = 0; t < 12; ++t) bf[t] = *(const v2f*)(bbase + t * 32);
#pragma unroll
      for (int t = 0; t < 12; ++t) {
        acc[t] = __builtin_amdgcn_wmma_f32_16x16x4_f32(
            /*neg_a=*/false, a, /*neg_b=*/false, bf[t],
            /*c_mod=*/(short)0, acc[t], /*reuse_a=*/false, /*reuse_b=*/false);
      }
    }
  }

  if (m0 >= nrows) return;
#pragma unroll
  for (int t = 0; t < 12; ++t) {
#pragma unroll
    for (int r = 0; r < 8; ++r) {
      const int rr = m0 + r + 8 * hi;  // C/D layout: VGPR r = rows M=r / M=r+8
      if (rr < nrows) Y[(size_t)rr * GH + t * 16 + lo] = acc[t][r];
    }
  }
}

// ---------------------------------------------------------------------------
// Edge aggregation: agg[dst] += w * support[src], one wave per edge,
// 6 floats per lane, hardware fp32 atomics (L2-resident).
// ---------------------------------------------------------------------------
__global__ __launch_bounds__(256) void scatter192(const float* __restrict__ sup,
                                                  const int* __restrict__ esrc,
                                                  const int* __restrict__ edst,
                                                  const float* __restrict__ ew,
                                                  float* __restrict__ agg, int ne) {
  const int e = blockIdx.x * 8 + (threadIdx.x >> 5);
  if (e >= ne) return;
  const int lane = threadIdx.x & 31;
  const int s = esrc[e];
  const int d = edst[e];
  const float wt = ew[e];
  const float* sp = sup + (size_t)s * GH;
  float* ap = agg + (size_t)d * GH;
#pragma unroll
  for (int j = 0; j < 6; ++j) {
    const int c = lane + 32 * j;
    __hip_atomic_fetch_add(ap + c, wt * sp[c], __ATOMIC_RELAXED,
                           __HIP_MEMORY_SCOPE_AGENT);
  }
}

__global__ void bias_relu_k(const float* __restrict__ agg,
                            const float* __restrict__ b,
                            float* __restrict__ out) {
  const size_t i = (size_t)blockIdx.x * GH + threadIdx.x;
  const float v = agg[i] + b[threadIdx.x];
  out[i] = fmaxf(v, 0.0f);
}

__global__ void residual_k(const float* __restrict__ a,
                           const float* __restrict__ x,
                           float* __restrict__ out, int n) {
  const int i = blockIdx.x * blockDim.x + threadIdx.x;
  if (i < n) out[i] = (a[i] + x[i]) * 0.5f;
}

// Head: sup3[N,3] = feats[N,192] @ W_out[192,3]
__global__ __launch_bounds__(256) void head_gemm(const float* __restrict__ F,
                                                 const float* __restrict__ Wo,
                                                 float* __restrict__ Y, int nrows) {
  const int row = blockIdx.x * blockDim.x + threadIdx.x;
  if (row >= nrows) return;
  const float* f = F + (size_t)row * GH;
  float a0 = 0.f, a1 = 0.f, a2 = 0.f;
#pragma unroll 4
  for (int k = 0; k < GH; ++k) {
    const float fv = f[k];
    a0 = fmaf(fv, Wo[k * 3 + 0], a0);
    a1 = fmaf(fv, Wo[k * 3 + 1], a1);
    a2 = fmaf(fv, Wo[k * 3 + 2], a2);
  }
  Y[(size_t)row * 3 + 0] = a0;
  Y[(size_t)row * 3 + 1] = a1;
  Y[(size_t)row * 3 + 2] = a2;
}

__global__ __launch_bounds__(256) void scatter3(const float* __restrict__ sup3,
                                                const int* __restrict__ esrc,
                                                const int* __restrict__ edst,
                                                const float* __restrict__ ew,
                                                float* __restrict__ coords, int ne) {
  const int e = blockIdx.x * blockDim.x + threadIdx.x;
  if (e >= ne) return;
  const int s = esrc[e];
  const int d = edst[e];
  const float wt = ew[e];
#pragma unroll
  for (int c = 0; c < 3; ++c) {
    __hip_atomic_fetch_add(coords + (size_t)d * 3 + c,
                           wt * sup3[(size_t)s * 3 + c], __ATOMIC_RELAXED,
                           __HIP_MEMORY_SCOPE_AGENT);
  }
}

__global__ void add_bout_k(float* __restrict__ coords,
                           const float* __restrict__ bout, int n3) {
  const int i = blockIdx.x * blockDim.x + threadIdx.x;
  if (i < n3) coords[i] += bout[i % 3];
}

extern "C" void kernel_launch(void* const* d_in, const int* in_sizes, int n_in,
                              void* d_out, int out_size, void* d_ws, size_t ws_size,
                              hipStream_t stream) {
  const float* features = (const float*)d_in[0];
  const int* esrc = (const int*)d_in[1];
  const int* edst = (const int*)d_in[2];
  const float* ew = (const float*)d_in[3];
  const float* Ws = (const float*)d_in[4];
  const float* bs = (const float*)d_in[5];
  const float* Wout = (const float*)d_in[6];
  const float* bout = (const float*)d_in[7];

  float* coords = (float*)d_out;                    // [N,3]
  float* featsOut = coords + (size_t)GN * 3;        // [N,192]

  const size_t NH = (size_t)GN * GH;
  float* feats = (float*)d_ws;
  float* xbuf = feats + NH;
  float* sup = xbuf + NH;
  float* agg = sup + NH;
  float* sup3 = agg + NH;                           // [N,3]

  const int gemmGrid = ((GN + 15) / 16 + 7) / 8;    // 196
  const int scatGrid = (GE + 7) / 8;                // 50000
  const int resGrid = (int)((NH + 255) / 256);

  auto gcn = [&](const float* in, int li, float* out) {
    gemm192_wmma<<<gemmGrid, 256, 0, stream>>>(in, Ws + (size_t)li * GH * GH, sup, GN);
    hipMemsetAsync(agg, 0, NH * sizeof(float), stream);
    scatter192<<<scatGrid, 256, 0, stream>>>(sup, esrc, edst, ew, agg, GE);
    bias_relu_k<<<GN, GH, 0, stream>>>(agg, bs + (size_t)li * GH, out);
  };

  // block 1
  gcn(features, 0, xbuf);
  gcn(xbuf, 1, xbuf);
  residual_k<<<resGrid, 256, 0, stream>>>(features, xbuf, feats, (int)NH);
  // blocks 2-6
  for (int i = 2; i < 12; i += 2) {
    gcn(feats, i, xbuf);
    gcn(xbuf, i + 1, xbuf);
    residual_k<<<resGrid, 256, 0, stream>>>(feats, xbuf, feats, (int)NH);
  }
  // gc13 + residual, writing final feats directly into d_out
  gcn(feats, 12, xbuf);
  residual_k<<<resGrid, 256, 0, stream>>>(feats, xbuf, featsOut, (int)NH);

  // output head (gc14)
  head_gemm<<<(GN + 255) / 256, 256, 0, stream>>>(featsOut, Wout, sup3, GN);
  hipMemsetAsync(coords, 0, (size_t)GN * 3 * sizeof(float), stream);
  scatter3<<<(GE + 255) / 256, 256, 0, stream>>>(sup3, esrc, edst, ew, coords, GE);
  add_bout_k<<<(GN * 3 + 255) / 256, 256, 0, stream>>>(coords, bout, GN * 3);
}